// QVLoraExpertRouter_42666205118700
// MI455X (gfx1250) — compile-verified
//
#include <hip/hip_runtime.h>
#include <hip/hip_bf16.h>

typedef __attribute__((ext_vector_type(2))) float v2f;
typedef __attribute__((ext_vector_type(8))) float v8f;

#define NCAT 320          // 256 lora_a cols + 8 router + 56 zero pad -> 20 n-tiles
#define LORA_SCALE 2.0f   // alpha/rank = 32/16

// ---------------------------------------------------------------------------
// Pack Wcat[I][320]: cols [e*32 .. e*32+15] = q_lora_a[e][i][r],
//                    cols [e*32+16 .. +31]  = v_lora_a[e][i][r],
//                    cols [256..263] = router_weight[e][i], cols [264..319]=0
// ---------------------------------------------------------------------------
__global__ void k_prep_wcat(const float* __restrict__ rw,
                            const float* __restrict__ qa,
                            const float* __restrict__ va,
                            float* __restrict__ Wcat, int I) {
  int idx = blockIdx.x * blockDim.x + threadIdx.x;
  int total = I * NCAT;
  if (idx >= total) return;
  int i = idx / NCAT, n = idx % NCAT;
  float v = 0.f;
  if (n < 256) {
    int e = n >> 5, w = n & 31;
    if (w < 16) v = qa[((size_t)e * I + i) * 16 + w];
    else        v = va[((size_t)e * I + i) * 16 + (w - 16)];
  } else if (n < 264) {
    v = rw[(size_t)(n - 256) * I + i];
  }
  Wcat[idx] = v;
}

// ---------------------------------------------------------------------------
// Fused router + low-rank GEMM.  Block = 16 tokens, 4 waves (wave32).
// P[16,320] = h_tile[16,K] @ Wcat[K,320] via V_WMMA_F32_16X16X4_F32,
// A chunks staged through LDS (coalesced HBM reads of h, read exactly once).
// Epilogue: softmax/argmax over logits (cols 256..263), write gated
// lowQ/lowV [T,128] (zero outside the selected expert).
// ---------------------------------------------------------------------------
__global__ __launch_bounds__(128) void k_router_lowrank(
    const float* __restrict__ h, const float* __restrict__ Wcat,
    float* __restrict__ gQ, float* __restrict__ gV, int I) {
  __shared__ float ldsA[16 * 68];     // 16 x 64 + pad
  __shared__ float ldsP[16 * NCAT];
  __shared__ float ldsGate[16];
  __shared__ int   ldsIdx[16];

  const int t = threadIdx.x;
  const int wave = t >> 5, lane = t & 31;
  const int m0 = blockIdx.x << 4;
  const int mlane = lane & 15;
  const int koff = (lane >> 4) << 1;   // K sub-offset per ISA A/B layout

  v8f acc[5] = {};                     // 5 n-tiles per wave (20 tiles / 4 waves)

  for (int k0 = 0; k0 < I; k0 += 64) {
    // cooperative, coalesced stage of A tile 16x64
    {
      int row = t >> 3;
      int kk = (t & 7) << 3;
      const float* src = h + (size_t)(m0 + row) * I + k0 + kk;
      float* dst = ldsA + row * 68 + kk;
#pragma unroll
      for (int j = 0; j < 8; ++j) dst[j] = src[j];
    }
    __syncthreads();
#pragma unroll 4
    for (int kk = 0; kk < 64; kk += 4) {
      v2f a;
      a.x = ldsA[mlane * 68 + kk + koff];
      a.y = ldsA[mlane * 68 + kk + koff + 1];
      int krow = k0 + kk + koff;
#pragma unroll
      for (int ti = 0; ti < 5; ++ti) {
        int n0 = (wave + (ti << 2)) << 4;
        v2f b;
        b.x = Wcat[(size_t)krow * NCAT + n0 + mlane];
        b.y = Wcat[(size_t)(krow + 1) * NCAT + n0 + mlane];
        acc[ti] = __builtin_amdgcn_wmma_f32_16x16x4_f32(
            false, a, false, b, (short)0, acc[ti], false, false);
      }
    }
    __syncthreads();
  }

  // C/D layout: VGPR i -> row i + (lane>>4)*8, col = lane&15
#pragma unroll
  for (int ti = 0; ti < 5; ++ti) {
    int n0 = (wave + (ti << 2)) << 4;
#pragma unroll
    for (int i = 0; i < 8; ++i) {
      int row = i + ((lane >> 4) << 3);
      ldsP[row * NCAT + n0 + mlane] = acc[ti][i];
    }
  }
  __syncthreads();

  if (t < 16) {                         // one thread per token: top-1 softmax
    const float* lg = ldsP + t * NCAT + 256;
    float m = lg[0]; int bi = 0;
    for (int e = 1; e < 8; ++e) { float x = lg[e]; if (x > m) { m = x; bi = e; } }
    float s = 0.f;
    for (int e = 0; e < 8; ++e) s += __expf(lg[e] - m);
    ldsGate[t] = LORA_SCALE / s;        // score = exp(0)/sum = 1/s
    ldsIdx[t] = bi;
  }
  __syncthreads();

  for (int i = t; i < 16 * 128; i += 128) {
    int row = i >> 7, col = i & 127;
    int e = col >> 4, r = col & 15;
    int bi = ldsIdx[row];
    float g = ldsGate[row];
    float vq = (e == bi) ? ldsP[row * NCAT + (bi << 5) + r] * g : 0.f;
    float vv = (e == bi) ? ldsP[row * NCAT + (bi << 5) + 16 + r] * g : 0.f;
    gQ[(size_t)(m0 + row) * 128 + col] = vq;
    gV[(size_t)(m0 + row) * 128 + col] = vv;
  }
}

// ---------------------------------------------------------------------------
// Dense expand: Out[T,N] = Alow[T,128] @ Bmat[128,N]   (N = 4096 or 1024)
// Block = 8 waves sharing one 16x128 A tile in LDS; each wave owns one
// 16x16 output tile, K=128 -> 32 WMMAs. Non-temporal output stores keep the
// 320 MB streaming writes from thrashing L2 (Alow/Bmat stay hot there).
// ---------------------------------------------------------------------------
__global__ __launch_bounds__(256) void k_expand(
    const float* __restrict__ Alow, const float* __restrict__ Bmat,
    float* __restrict__ Out, int Ncols) {
  __shared__ float ldsA[16 * 132];     // 16 x 128 + pad
  const int t = threadIdx.x;
  const int wave = t >> 5, lane = t & 31;
  const int m0 = blockIdx.x << 4;
  const int n0 = ((blockIdx.y << 3) + wave) << 4;

  {
    int row = t >> 4;
    int kk = (t & 15) << 3;
    const float* src = Alow + (size_t)(m0 + row) * 128 + kk;
    float* dst = ldsA + row * 132 + kk;
#pragma unroll
    for (int j = 0; j < 8; ++j) dst[j] = src[j];
  }
  __syncthreads();

  const int mlane = lane & 15;
  const int koff = (lane >> 4) << 1;
  v8f acc = {};
#pragma unroll
  for (int k0 = 0; k0 < 128; k0 += 4) {
    v2f a;
    a.x = ldsA[mlane * 132 + k0 + koff];
    a.y = ldsA[mlane * 132 + k0 + koff + 1];
    v2f b;
    b.x = Bmat[(size_t)(k0 + koff) * Ncols + n0 + mlane];
    b.y = Bmat[(size_t)(k0 + koff + 1) * Ncols + n0 + mlane];
    acc = __builtin_amdgcn_wmma_f32_16x16x4_f32(
        false, a, false, b, (short)0, acc, false, false);
  }
#pragma unroll
  for (int i = 0; i < 8; ++i) {
    int row = m0 + i + ((lane >> 4) << 3);
    __builtin_nontemporal_store(acc[i], Out + (size_t)row * Ncols + n0 + mlane);
  }
}

// ---------------------------------------------------------------------------
extern "C" void kernel_launch(void* const* d_in, const int* in_sizes, int n_in,
                              void* d_out, int out_size, void* d_ws, size_t ws_size,
                              hipStream_t stream) {
  const float* h  = (const float*)d_in[0];
  const float* rw = (const float*)d_in[1];
  const float* qa = (const float*)d_in[2];
  const float* qb = (const float*)d_in[3];
  const float* va = (const float*)d_in[4];
  const float* vb = (const float*)d_in[5];

  const int E = 8, R = 16;
  const int I = in_sizes[1] / E;                 // 4096
  const int T = in_sizes[0] / I;                 // 16384
  const int Q = in_sizes[3] / (E * R);           // 4096
  const int V = in_sizes[5] / (E * R);           // 1024

  float* ws   = (float*)d_ws;
  float* Wcat = ws;                              // I * 320 floats
  float* gQ   = Wcat + (size_t)I * NCAT;         // T * 128 floats
  float* gV   = gQ + (size_t)T * 128;            // T * 128 floats
  float* out  = (float*)d_out;

  {
    int total = I * NCAT;
    k_prep_wcat<<<(total + 255) / 256, 256, 0, stream>>>(rw, qa, va, Wcat, I);
  }
  k_router_lowrank<<<T / 16, 128, 0, stream>>>(h, Wcat, gQ, gV, I);
  k_expand<<<dim3(T / 16, Q / 128), 256, 0, stream>>>(gQ, qb, out, Q);
  k_expand<<<dim3(T / 16, V / 128), 256, 0, stream>>>(gV, vb, out + (size_t)T * Q, V);
}